// VertexNormals_18622978196254
// MI455X (gfx1250) — compile-verified
//
#include <hip/hip_runtime.h>
#include <math.h>

#define GS    512
#define NVERT (GS*GS)
#define TW    32          // tile width (vertices)
#define THT   8           // tile height (vertices)
#define VROWS (THT+2)     // 10 halo rows
#define VCOLS (TW+2)      // 34 halo cols
#define QROWS (THT+1)     // 9 quad rows
#define QCOLS (TW+1)      // 33 quad cols
#define TILE_DWORDS (VROWS*VCOLS*3)   // 1020

typedef unsigned int       u32;
typedef unsigned long long u64;
typedef u32 u32x4 __attribute__((ext_vector_type(4)));
typedef u32 u32x8 __attribute__((ext_vector_type(8)));

__global__ __launch_bounds__(256)
void vertex_normals_stencil(const float* __restrict__ vrt, float* __restrict__ out)
{
    __shared__ float sv[VROWS*VCOLS*3];   // halo vertex tile   (4080 B)
    __shared__ float sn[QROWS*QCOLS*6];   // 2 normals per quad (7128 B)

    const int bx  = blockIdx.x;           // 0..15  (column tiles)
    const int by  = blockIdx.y;           // 0..63  (row tiles)
    const int b   = blockIdx.z;           // batch
    const int tid = threadIdx.x;
    const int r0  = by * THT;
    const int c0  = bx * TW;

    // LDS byte offset of sv within the workgroup's LDS allocation
    const u32 sv_off = (u32)(size_t)(__attribute__((address_space(3))) float*)&sv[0];
    const u64 gbase  = (u64)(size_t)(vrt + (size_t)b * NVERT * 3);

    const bool interior = (by >= 1) & (by <= 62) & (bx >= 1) & (bx <= 14);

    if (interior) {
        // ---- Tensor Data Mover: one 2D tile DMA (102 dwords x 10 rows) into LDS ----
        if ((tid >> 5) == 0) {            // wave 0 only (EXEC ignored by TDM; uniform branch)
            u64 ga = gbase + 4ull * (((u64)(r0 - 1) * GS + (u64)(c0 - 1)) * 3);
            // D# group 0: count=1 | lds_addr | global_addr[56:0] | type=2
            u32x4 g0 = { 1u,
                         sv_off,
                         (u32)(ga & 0xffffffffu),
                         (u32)((ga >> 32) & 0x01ffffffu) | (2u << 30) };
            // D# group 1
            const u32 dim0 = GS * 3;      // 1536 elements per tensor row
            u32x8 g1 = { (2u << 16),                        // data_size=4B, mask=0
                         (dim0 & 0xffffu) << 16,            // tensor_dim0 lo16
                         (dim0 >> 16) | ((u32)GS << 16),    // dim0 hi | tensor_dim1 lo16
                         ((u32)GS >> 16) | ((u32)(VCOLS*3) << 16), // dim1 hi | tile_dim0=102
                         (u32)VROWS,                        // tile_dim1=10, tile_dim2=0
                         dim0,                              // tensor_dim0_stride lo32
                         0u,                                // stride hi | dim1_stride lo
                         0u };
            asm volatile("tensor_load_to_lds %0, %1" :: "s"(g0), "s"(g1) : "memory");
            __builtin_amdgcn_s_wait_tensorcnt(0);
        }
    } else {
        // ---- Boundary tiles: clamped per-lane async loads straight into LDS ----
        for (int d = tid; d < TILE_DWORDS; d += 256) {
            int i  = d / (VCOLS * 3);
            int j  = d - i * (VCOLS * 3);
            int cc = j / 3;
            int k  = j - cc * 3;
            int gr = r0 - 1 + i;  gr = gr < 0 ? 0 : (gr > GS - 1 ? GS - 1 : gr);
            int gc = c0 - 1 + cc; gc = gc < 0 ? 0 : (gc > GS - 1 ? GS - 1 : gc);
            u64 ga   = gbase + 4ull * (((u64)gr * GS + (u64)gc) * 3 + (u64)k);
            u32 ldso = sv_off + (u32)d * 4u;
            asm volatile("global_load_async_to_lds_b32 %0, %1, off"
                         :: "v"(ldso), "v"(ga) : "memory");
        }
        asm volatile("s_wait_asynccnt 0x0" ::: "memory");
    }
    __syncthreads();

    // ---- Phase 1: unit face normals for all 9x33 quads (2 triangles each) ----
    for (int q = tid; q < QROWS * QCOLS; q += 256) {
        int li  = q / QCOLS;
        int lj  = q - li * QCOLS;
        int i00 = (li * VCOLS + lj) * 3;
        int i01 = i00 + 3;
        int i10 = ((li + 1) * VCOLS + lj) * 3;
        int i11 = i10 + 3;
        // t1 = [v00, v10, v01] : cross(v10-v00, v01-v00)
        float ax = sv[i10+0]-sv[i00+0], ay = sv[i10+1]-sv[i00+1], az = sv[i10+2]-sv[i00+2];
        float bx_= sv[i01+0]-sv[i00+0], by_= sv[i01+1]-sv[i00+1], bz = sv[i01+2]-sv[i00+2];
        float n1x = ay*bz - az*by_;
        float n1y = az*bx_ - ax*bz;
        float n1z = ax*by_ - ay*bx_;
        float l1  = fmaxf(sqrtf(n1x*n1x + n1y*n1y + n1z*n1z), 1e-12f);
        // t2 = [v01, v10, v11] : cross(v10-v01, v11-v01)
        float cx = sv[i10+0]-sv[i01+0], cy = sv[i10+1]-sv[i01+1], cz = sv[i10+2]-sv[i01+2];
        float dx = sv[i11+0]-sv[i01+0], dy = sv[i11+1]-sv[i01+1], dz = sv[i11+2]-sv[i01+2];
        float n2x = cy*dz - cz*dy;
        float n2y = cz*dx - cx*dz;
        float n2z = cx*dy - cy*dx;
        float l2  = fmaxf(sqrtf(n2x*n2x + n2y*n2y + n2z*n2z), 1e-12f);
        int o = q * 6;
        sn[o+0] = n1x/l1; sn[o+1] = n1y/l1; sn[o+2] = n1z/l1;
        sn[o+3] = n2x/l2; sn[o+4] = n2y/l2; sn[o+5] = n2z/l2;
    }
    __syncthreads();

    // ---- Phase 2: per-vertex accumulation over incident triangles ----
    const int ri = tid >> 5;          // 0..7
    const int ci = tid & 31;          // 0..31
    const int r  = r0 + ri;
    const int c  = c0 + ci;

    float axv = 0.f, ayv = 0.f, azv = 0.f;
    if (r < GS-1 && c < GS-1) {                       // t1 of quad (r, c)
        int o = ((ri+1)*QCOLS + (ci+1)) * 6;
        axv += sn[o+0]; ayv += sn[o+1]; azv += sn[o+2];
    }
    if (r >= 1 && c < GS-1) {                         // t1+t2 of quad (r-1, c)
        int o = (ri*QCOLS + (ci+1)) * 6;
        axv += sn[o+0] + sn[o+3]; ayv += sn[o+1] + sn[o+4]; azv += sn[o+2] + sn[o+5];
    }
    if (r < GS-1 && c >= 1) {                         // t1+t2 of quad (r, c-1)
        int o = ((ri+1)*QCOLS + ci) * 6;
        axv += sn[o+0] + sn[o+3]; ayv += sn[o+1] + sn[o+4]; azv += sn[o+2] + sn[o+5];
    }
    if (r >= 1 && c >= 1) {                           // t2 of quad (r-1, c-1)
        int o = (ri*QCOLS + ci) * 6;
        axv += sn[o+3]; ayv += sn[o+4]; azv += sn[o+5];
    }
    float l = fmaxf(sqrtf(axv*axv + ayv*ayv + azv*azv), 1e-12f);
    size_t oi = ((size_t)b * NVERT + (size_t)r * GS + (size_t)c) * 3;
    out[oi+0] = axv / l;
    out[oi+1] = ayv / l;
    out[oi+2] = azv / l;
}

extern "C" void kernel_launch(void* const* d_in, const int* in_sizes, int n_in,
                              void* d_out, int out_size, void* d_ws, size_t ws_size,
                              hipStream_t stream)
{
    const float* vrt = (const float*)d_in[0];
    float* out = (float*)d_out;
    int bs = in_sizes[0] / (NVERT * 3);               // = 16
    dim3 grid(GS / TW, GS / THT, bs);                  // (16, 64, 16)
    vertex_normals_stencil<<<grid, 256, 0, stream>>>(vrt, out);
    (void)n_in; (void)out_size; (void)d_ws; (void)ws_size;
}